// HGRMultiCase_10754598109733
// MI455X (gfx1250) — compile-verified
//
#include <hip/hip_runtime.h>

// HGR multi-case: KNN hypergraph + 2x HGNN conv + pool + FC + sigmoid
// Target: gfx1250 (MI455X), wave32, f32 WMMA 16x16x4 + async global->LDS staging.

typedef __attribute__((ext_vector_type(2))) float v2f;
typedef __attribute__((ext_vector_type(8))) float v8f;

#define N_NODES 8192
#define IN_CH   1024
#define HID     128
#define KNN     10
#define SLOPE   0.01f

#define JT_COLS 32      // columns processed per j-iteration (2 j-tiles/wave)
#define KC      128     // K chunk staged in LDS
#define BSTRIDE 132     // padded LDS row stride (floats): 132%64==4 -> no bank conflicts

__device__ __forceinline__ v8f wmma_f32_16x16x4(v2f a, v2f b, v8f c) {
  // D = A(16x4) * B(4x16) + C, fp32 WMMA
  return __builtin_amdgcn_wmma_f32_16x16x4_f32(false, a, false, b, (short)0, c,
                                               false, false);
}

// ---------------------------------------------------------------- row norms
__global__ __launch_bounds__(256) void k_sqnorm(const float* __restrict__ x,
                                                float* __restrict__ sq) {
  __shared__ float red[256];
  const int row = blockIdx.x;
  const int t = threadIdx.x;
  float s = 0.f;
  for (int c = t; c < IN_CH; c += 256) {
    float v = x[(size_t)row * IN_CH + c];
    s += v * v;
  }
  red[t] = s;
  __syncthreads();
  for (int off = 128; off > 0; off >>= 1) {
    if (t < off) red[t] += red[t + off];
    __syncthreads();
  }
  if (t == 0) sq[row] = red[0];
}

// -------------------------------------------- fused Gram + distance + top-K
// Block = 4 waves, each owns a 16-row i-tile. All waves share a 32-column
// B tile staged in LDS via GLOBAL_LOAD_ASYNC_TO_LDS_B128 (ASYNCcnt), and each
// wave register-blocks 2 j-tiles: per 2 WMMAs -> 1 global b64 + 2 ds b64.
__global__ __launch_bounds__(128) void k_dist_topk(const float* __restrict__ x,
                                                   const float* __restrict__ sq,
                                                   int* __restrict__ nnidx,
                                                   int* __restrict__ Dv) {
  __shared__ float stage[32 * BSTRIDE];   // 16.9 KB staged B chunk (32 x KC)
  __shared__ float tile[4][16][34];       // per-wave 16x32 distance tile (padded)
  __shared__ float md[4][32][KNN];
  __shared__ int   mi[4][32][KNN];

  const int tid  = threadIdx.x;
  const int wave = tid >> 5;
  const int lane = tid & 31;
  const int iBase = (blockIdx.x * 4 + wave) * 16;
  const int m    = lane & 15;              // A row-in-tile / B column
  const int half = lane >> 4;
  const int koff = half * 2;               // K pair offset per ISA layout

  float bd[KNN];
  int   bi[KNN];
#pragma unroll
  for (int t = 0; t < KNN; ++t) { bd[t] = 3.0e38f; bi[t] = 0; }

  const float* arow = x + (size_t)(iBase + m) * IN_CH + koff;
  float sqi[8];
#pragma unroll
  for (int r = 0; r < 8; ++r) sqi[r] = sq[iBase + r + 8 * half];

  // cooperative staging: thread -> (row 0..31, 32-float slab)
  const int srow = tid >> 2;
  const int scol = (tid & 3) * 32;
  const unsigned ldsStageBase =
      (unsigned)(uintptr_t)(&stage[srow * BSTRIDE + scol]);  // low 32b = LDS offset

  // per-lane LDS read pointers for the two B fragments
  const float* b0p = &stage[m * BSTRIDE + koff];
  const float* b1p = &stage[(m + 16) * BSTRIDE + koff];

  for (int jt = 0; jt < N_NODES / JT_COLS; ++jt) {
    const int jBase = jt * JT_COLS;
    v8f acc0 = {};
    v8f acc1 = {};
    const float* gsrc = x + (size_t)(jBase + srow) * IN_CH + scol;

    for (int kc = 0; kc < IN_CH; kc += KC) {
      __syncthreads();  // previous chunk fully consumed by all waves
#pragma unroll
      for (int i = 0; i < 8; ++i) {
        const unsigned ldsa = ldsStageBase + (unsigned)(i * 16);
        const unsigned long long ga =
            (unsigned long long)(uintptr_t)(gsrc + kc + i * 4);
        asm volatile("global_load_async_to_lds_b128 %0, %1, off"
                     :: "v"(ldsa), "v"(ga)
                     : "memory");
      }
      asm volatile("s_wait_asynccnt 0" ::: "memory");
      __syncthreads();  // staged chunk visible to all 4 waves

      for (int k = 0; k < KC; k += 4) {
        v2f a  = *(const v2f*)(arow + kc + k);   // 8B-aligned b64
        v2f b0 = *(const v2f*)(b0p + k);         // conflict-free ds b64
        v2f b1 = *(const v2f*)(b1p + k);
        acc0 = wmma_f32_16x16x4(a, b0, acc0);
        acc1 = wmma_f32_16x16x4(a, b1, acc1);
      }
    }

    // distance epilogue -> LDS (transpose to row-per-lane for top-K scan)
    const float sqj0 = sq[jBase + m];
    const float sqj1 = sq[jBase + 16 + m];
#pragma unroll
    for (int r = 0; r < 8; ++r) {
      const int mm = r + 8 * half;       // C/D layout: VGPR r, hi lanes -> M+8
      tile[wave][mm][m]      = sqi[r] + sqj0 - 2.0f * acc0[r];
      tile[wave][mm][16 + m] = sqi[r] + sqj1 - 2.0f * acc1[r];
    }
    __builtin_amdgcn_wave_barrier();     // keep per-wave LDS store->load in order
#pragma unroll
    for (int c = 0; c < 16; ++c) {
      const int col = half * 16 + c;     // 2 lanes per row, 16 cols each
      float d = tile[wave][m][col];
      int idx = jBase + col;
      if (d < bd[KNN - 1]) {             // early-out: rare after warm-up
#pragma unroll
        for (int s = 0; s < KNN; ++s) {
          if (d < bd[s]) {
            float td = bd[s]; int ti = bi[s];
            bd[s] = d; bi[s] = idx; d = td; idx = ti;
          }
        }
      }
    }
    __builtin_amdgcn_wave_barrier();
  }

  // merge the two half-lane lists of each row
#pragma unroll
  for (int t = 0; t < KNN; ++t) { md[wave][lane][t] = bd[t]; mi[wave][lane][t] = bi[t]; }
  __builtin_amdgcn_wave_barrier();
  if (lane < 16) {
#pragma unroll
    for (int t = 0; t < KNN; ++t) {
      float d = md[wave][lane + 16][t];
      int idx = mi[wave][lane + 16][t];
      if (d < bd[KNN - 1]) {
#pragma unroll
        for (int s = 0; s < KNN; ++s) {
          if (d < bd[s]) {
            float td = bd[s]; int ti = bi[s];
            bd[s] = d; bi[s] = idx; d = td; idx = ti;
          }
        }
      }
    }
    const int row = iBase + lane;
#pragma unroll
    for (int t = 0; t < KNN; ++t) {
      nnidx[row * KNN + t] = bi[t];
      atomicAdd(&Dv[bi[t]], 1);          // node degree
    }
  }
}

// ----------------------------------------------------- generic f32 WMMA GEMM
// C[M x Nout] = A[M x Kdim] * B[Kdim x Nout], one 16x16 tile per wave.
__global__ __launch_bounds__(128) void k_gemm_f32(const float* __restrict__ A,
                                                  const float* __restrict__ B,
                                                  float* __restrict__ C,
                                                  int Kdim, int Nout) {
  const int wave = threadIdx.x >> 5;
  const int lane = threadIdx.x & 31;
  const int tilesN = Nout >> 4;
  const int tileId = blockIdx.x * 4 + wave;
  const int iBase = (tileId / tilesN) * 16;
  const int jBase = (tileId % tilesN) * 16;
  const int m = lane & 15;
  const int half = lane >> 4;
  const int koff = half * 2;

  const float* arow = A + (size_t)(iBase + m) * Kdim + koff;
  const float* bcol = B + jBase + m;
  v8f acc = {};
  for (int k = 0; k < Kdim; k += 4) {
    v2f a = *(const v2f*)(arow + k);
    v2f b;
    b.x = bcol[(size_t)(k + koff) * Nout];
    b.y = bcol[(size_t)(k + koff + 1) * Nout];
    acc = wmma_f32_16x16x4(a, b, acc);
  }
#pragma unroll
  for (int r = 0; r < 8; ++r)
    C[(size_t)(iBase + r + 8 * half) * Nout + jBase + m] = acc[r];
}

// ------------------------------------------- node -> hyperedge (mean over K)
__global__ __launch_bounds__(HID) void k_edge_mean(const float* __restrict__ y,
                                                   const int* __restrict__ nnidx,
                                                   float* __restrict__ eft) {
  __shared__ int nb[KNN];
  const int e = blockIdx.x, c = threadIdx.x;
  if (c < KNN) nb[c] = nnidx[e * KNN + c];
  __syncthreads();
  float s = 0.f;
#pragma unroll
  for (int j = 0; j < KNN; ++j) s += y[(size_t)nb[j] * HID + c];
  eft[(size_t)e * HID + c] = s * (1.0f / KNN);
}

// ------------------------------------------- hyperedge -> node (scatter-add)
__global__ __launch_bounds__(HID) void k_scatter(const float* __restrict__ eft,
                                                 const int* __restrict__ nnidx,
                                                 float* __restrict__ nft) {
  __shared__ int nb[KNN];
  const int e = blockIdx.x, c = threadIdx.x;
  if (c < KNN) nb[c] = nnidx[e * KNN + c];
  __syncthreads();
  const float v = eft[(size_t)e * HID + c];
#pragma unroll
  for (int j = 0; j < KNN; ++j)
    atomicAdd(&nft[(size_t)nb[j] * HID + c], v);
}

// ----------------------------------------- /max(Dv,1) + bias + leaky_relu
__global__ __launch_bounds__(256) void k_finalize(const float* __restrict__ nft,
                                                  const int* __restrict__ Dv,
                                                  const float* __restrict__ bias,
                                                  float* __restrict__ out) {
  const int idx = blockIdx.x * 256 + threadIdx.x;
  const int v = idx >> 7, c = idx & (HID - 1);
  const int dv = Dv[v];
  const float inv = 1.0f / (float)(dv < 1 ? 1 : dv);
  const float h = nft[idx] * inv + bias[c];
  out[idx] = h >= 0.f ? h : SLOPE * h;
}

// ------------------------------------------------ column mean (partial+atomic)
__global__ __launch_bounds__(HID) void k_colmean(const float* __restrict__ feats,
                                                 float* __restrict__ pool) {
  const int c = threadIdx.x;
  const int v0 = blockIdx.x * 128;
  float s = 0.f;
  for (int v = 0; v < 128; ++v) s += feats[(size_t)(v0 + v) * HID + c];
  atomicAdd(&pool[c], s * (1.0f / N_NODES));
}

// -------------------------------------------------------- FC(2) + sigmoid
__global__ __launch_bounds__(HID) void k_logits(const float* __restrict__ pool,
                                                const float* __restrict__ fcw,
                                                const float* __restrict__ fcb,
                                                float* __restrict__ outSig) {
  __shared__ float red[HID];
  const int c = threadIdx.x;
  const float p = pool[c];
  for (int o = 0; o < 2; ++o) {
    red[c] = fcw[o * HID + c] * p;
    __syncthreads();
    for (int off = 64; off > 0; off >>= 1) {
      if (c < off) red[c] += red[c + off];
      __syncthreads();
    }
    if (c == 0) {
      const float z = red[0] + fcb[o];
      outSig[o] = 1.0f / (1.0f + expf(-z));
    }
    __syncthreads();
  }
}

extern "C" void kernel_launch(void* const* d_in, const int* in_sizes, int n_in,
                              void* d_out, int out_size, void* d_ws, size_t ws_size,
                              hipStream_t stream) {
  (void)in_sizes; (void)n_in; (void)out_size; (void)ws_size;

  const float* x      = (const float*)d_in[0];
  const float* theta0 = (const float*)d_in[1];
  const float* bias0  = (const float*)d_in[2];
  const float* theta1 = (const float*)d_in[3];
  const float* bias1  = (const float*)d_in[4];
  const float* fcw    = (const float*)d_in[5];
  const float* fcb    = (const float*)d_in[6];

  float* out   = (float*)d_out;
  float* feats = out + 2;                               // [8192,128]
  float* pool  = out + 2 + (size_t)N_NODES * HID;       // [128]

  // workspace layout (~17.2 MB)
  float* ws    = (float*)d_ws;
  float* sq    = ws;                                    // 8192 f32
  int*   nnidx = (int*)(ws + N_NODES);                  // 81920 i32
  int*   Dv    = (int*)(ws + N_NODES + N_NODES * KNN);  // 8192 i32
  float* y     = ws + N_NODES + N_NODES * KNN + N_NODES;
  float* eft   = y   + (size_t)N_NODES * HID;
  float* nft   = eft + (size_t)N_NODES * HID;
  float* h1    = nft + (size_t)N_NODES * HID;

  // graph build
  hipMemsetAsync(Dv, 0, N_NODES * sizeof(int), stream);
  k_sqnorm<<<N_NODES, 256, 0, stream>>>(x, sq);
  k_dist_topk<<<N_NODES / 64, 128, 0, stream>>>(x, sq, nnidx, Dv);

  const int gemmBlocks = (N_NODES / 16) * (HID / 16) / 4;

  // layer 1
  k_gemm_f32<<<gemmBlocks, 128, 0, stream>>>(x, theta0, y, IN_CH, HID);
  k_edge_mean<<<N_NODES, HID, 0, stream>>>(y, nnidx, eft);
  hipMemsetAsync(nft, 0, (size_t)N_NODES * HID * sizeof(float), stream);
  k_scatter<<<N_NODES, HID, 0, stream>>>(eft, nnidx, nft);
  k_finalize<<<(N_NODES * HID) / 256, 256, 0, stream>>>(nft, Dv, bias0, h1);

  // layer 2
  k_gemm_f32<<<gemmBlocks, 128, 0, stream>>>(h1, theta1, y, HID, HID);
  k_edge_mean<<<N_NODES, HID, 0, stream>>>(y, nnidx, eft);
  hipMemsetAsync(nft, 0, (size_t)N_NODES * HID * sizeof(float), stream);
  k_scatter<<<N_NODES, HID, 0, stream>>>(eft, nnidx, nft);
  k_finalize<<<(N_NODES * HID) / 256, 256, 0, stream>>>(nft, Dv, bias1, feats);

  // pool + head
  hipMemsetAsync(pool, 0, HID * sizeof(float), stream);
  k_colmean<<<64, HID, 0, stream>>>(feats, pool);
  k_logits<<<1, HID, 0, stream>>>(pool, fcw, fcb, out);
}